// GLULayer_22342419874273
// MI455X (gfx1250) — compile-verified
//
#include <hip/hip_runtime.h>
#include <hip/hip_bf16.h>
#include <math.h>

typedef float v2f __attribute__((ext_vector_type(2)));
typedef float v8f __attribute__((ext_vector_type(8)));

// Problem sizes (fixed by the reference)
#define BN 4
#define SS 2048
#define DD 1024
#define HH 1024
#define OO 1024
#define FF 2048   // 2H

// ---------------------------------------------------------------------------
// Small GEMM: Out[m][n] = sum_k A[m][k] * B[k][n] (+bias[n]), m in [0, rows),
// rows <= 8, via V_WMMA_F32_16X16X4_F32.
// A: [16 x K] logical, row stride lda; lanes with m >= rows use a clamped
// pointer + multiplicative 0-mask (no divergence -> EXEC stays all-1s, as
// required by the WMMA ISA).  B: row-major [K x N] (TRANSB=0) or B[n][k]
// (TRANSB=1).  Block = 128 threads (4 waves); each wave owns K/4, K-loop
// unrolled x2 with two independent accumulators; LDS reduction at the end.
// ---------------------------------------------------------------------------
template <bool TRANSB>
__global__ void wmma_gemm16(const float* __restrict__ A, long lda, int rows,
                            const float* __restrict__ Bm, int ldb,
                            int K, const float* __restrict__ bias,
                            float* __restrict__ Out, int ldout) {
  __shared__ float red[4][32][8];
  const int lane = threadIdx.x & 31;
  const int wave = threadIdx.x >> 5;
  const int n0   = blockIdx.x * 16;
  const int ln   = lane & 15;            // M index (A frag), N index (B frag)
  const int kk   = (lane >> 4) * 2;      // lane-group K split {0,1} vs {2,3}
  const int kPerWave = K >> 2;
  const int kBeg = wave * kPerWave;

  // Branch-free A access: dead lanes read row 0 and multiply by 0.
  const float amask = (ln < rows) ? 1.f : 0.f;
  const float* aptr = A + (long)(ln < rows ? ln : 0) * lda + kBeg + kk;
  const float* bptr = TRANSB ? (Bm + (long)(n0 + ln) * ldb + kBeg + kk)
                             : (Bm + (long)(kBeg + kk) * ldb + (n0 + ln));
  const long brow  = TRANSB ? 1 : (long)ldb;        // b.x -> b.y offset
  const long bstep = TRANSB ? 4 : (long)4 * ldb;    // k -> k+4 offset

  v8f acc0 = {0.f, 0.f, 0.f, 0.f, 0.f, 0.f, 0.f, 0.f};
  v8f acc1 = {0.f, 0.f, 0.f, 0.f, 0.f, 0.f, 0.f, 0.f};

  const int iters = kPerWave >> 2;       // # of 16x16x4 steps for this wave
  for (int i = 0; i < iters; i += 2) {
    v2f a0, b0, a1, b1;
    a0.x = aptr[0] * amask;  a0.y = aptr[1] * amask;
    b0.x = bptr[0];          b0.y = bptr[brow];
    a1.x = aptr[4] * amask;  a1.y = aptr[5] * amask;
    b1.x = bptr[bstep];      b1.y = bptr[bstep + brow];
    acc0 = __builtin_amdgcn_wmma_f32_16x16x4_f32(
        false, a0, false, b0, (short)0, acc0, false, false);
    acc1 = __builtin_amdgcn_wmma_f32_16x16x4_f32(
        false, a1, false, b1, (short)0, acc1, false, false);
    aptr += 8;
    bptr += 2 * bstep;
  }

  for (int i = 0; i < 8; ++i) red[wave][lane][i] = acc0[i] + acc1[i];
  __syncthreads();

  // C/D layout: lanes 0-15 hold rows 0..7 in vgprs 0..7 (rows <= 8).
  if (wave == 0 && lane < 16) {
    for (int i = 0; i < rows; ++i) {
      float v = red[0][lane][i] + red[1][lane][i] +
                red[2][lane][i] + red[3][lane][i];
      if (bias) v += bias[n0 + lane];
      Out[(long)i * ldout + n0 + lane] = v;
    }
  }
}

// ---------------------------------------------------------------------------
// scores[b][s] = (qtil[b] . x[b][s]) / sqrt(1024).  One wave per row s.
// Streams x once (33.5 MB) at full bandwidth; qtil (4 KB/batch) stays cached.
// ---------------------------------------------------------------------------
__global__ void scores_k(const float* __restrict__ x,
                         const float* __restrict__ qtil,
                         float* __restrict__ scores) {
  const int b    = blockIdx.y;
  const int srow = blockIdx.x * 8 + (threadIdx.x >> 5);
  const int lane = threadIdx.x & 31;
  const float4* xr = (const float4*)(x + ((size_t)b * SS + srow) * DD);
  const float4* q4 = (const float4*)(qtil + (size_t)b * DD);
  float sum = 0.f;
#pragma unroll
  for (int i = 0; i < 8; ++i) {
    float4 xv = xr[lane + i * 32];
    float4 qv = q4[lane + i * 32];
    sum += xv.x * qv.x + xv.y * qv.y + xv.z * qv.z + xv.w * qv.w;
  }
  for (int off = 16; off; off >>= 1) sum += __shfl_xor(sum, off, 32);
  if (lane == 0) scores[b * SS + srow] = sum * 0.03125f;  // 1/sqrt(1024)
}

// ---------------------------------------------------------------------------
// Per-batch softmax over 2048 scores (in place); also zeroes cvec[b][:].
// ---------------------------------------------------------------------------
__global__ void softmax_zero(float* __restrict__ scores,
                             float* __restrict__ cvec) {
  const int b = blockIdx.x, tid = threadIdx.x;
  __shared__ float red[256];
  float* sc = scores + b * SS;
  float v[8];
  float mx = -3.402823466e38f;
#pragma unroll
  for (int j = 0; j < 8; ++j) { v[j] = sc[tid + j * 256]; mx = fmaxf(mx, v[j]); }
  red[tid] = mx; __syncthreads();
  for (int s = 128; s; s >>= 1) {
    if (tid < s) red[tid] = fmaxf(red[tid], red[tid + s]);
    __syncthreads();
  }
  mx = red[0]; __syncthreads();
  float sum = 0.f;
#pragma unroll
  for (int j = 0; j < 8; ++j) { v[j] = expf(v[j] - mx); sum += v[j]; }
  red[tid] = sum; __syncthreads();
  for (int s = 128; s; s >>= 1) {
    if (tid < s) red[tid] += red[tid + s];
    __syncthreads();
  }
  const float inv = 1.f / red[0];
#pragma unroll
  for (int j = 0; j < 8; ++j) sc[tid + j * 256] = v[j] * inv;
#pragma unroll
  for (int c = 0; c < 4; ++c) cvec[b * DD + tid * 4 + c] = 0.f;
}

// ---------------------------------------------------------------------------
// cvec[b][:] += sum over this block's s-chunk of w[b][s] * x[b][s][:].
// Second pass over x: x is L2-resident (33.5 MB << 192 MB) after scores_k.
// ---------------------------------------------------------------------------
__global__ void wsum_k(const float* __restrict__ x,
                       const float* __restrict__ wts,
                       float* __restrict__ cvec) {
  const int chunk = blockIdx.x, b = blockIdx.y, tid = threadIdx.x;
  const int s0 = chunk * 128;
  const float4* x4 = (const float4*)x;
  const float* w = wts + b * SS + s0;
  const size_t base = ((size_t)b * SS + s0) * (DD / 4);
  float4 acc = {0.f, 0.f, 0.f, 0.f};
  for (int s = 0; s < 128; ++s) {
    const float ww = w[s];
    float4 xv = x4[base + (size_t)s * (DD / 4) + tid];
    acc.x += ww * xv.x; acc.y += ww * xv.y;
    acc.z += ww * xv.z; acc.w += ww * xv.w;
  }
  float* c = cvec + b * DD + tid * 4;
  atomicAdd(c + 0, acc.x); atomicAdd(c + 1, acc.y);
  atomicAdd(c + 2, acc.z); atomicAdd(c + 3, acc.w);
}

// ---------------------------------------------------------------------------
// GLU (x1 * exact-gelu(x2)) + LayerNorm + residual with attn -> comb[b][:]
// ---------------------------------------------------------------------------
__global__ void glu_ln(const float* __restrict__ hin,
                       const float* __restrict__ attn,
                       const float* __restrict__ gamma,
                       const float* __restrict__ beta,
                       float* __restrict__ comb) {
  const int b = blockIdx.x, tid = threadIdx.x;
  __shared__ float red[256];
  const float* hb = hin + (size_t)b * FF;
  float hv[4];
  float sum = 0.f;
#pragma unroll
  for (int j = 0; j < 4; ++j) {
    const int i = tid + j * 256;
    const float x1 = hb[i];
    const float x2 = hb[HH + i];
    const float g = 0.5f * x2 * (1.f + erff(x2 * 0.70710678118654752f));
    hv[j] = x1 * g;
    sum += hv[j];
  }
  red[tid] = sum; __syncthreads();
  for (int s = 128; s; s >>= 1) { if (tid < s) red[tid] += red[tid + s]; __syncthreads(); }
  const float mean = red[0] * (1.f / HH); __syncthreads();
  float sq = 0.f;
#pragma unroll
  for (int j = 0; j < 4; ++j) { const float d = hv[j] - mean; sq += d * d; }
  red[tid] = sq; __syncthreads();
  for (int s = 128; s; s >>= 1) { if (tid < s) red[tid] += red[tid + s]; __syncthreads(); }
  const float rstd = rsqrtf(red[0] * (1.f / HH) + 1e-5f);
#pragma unroll
  for (int j = 0; j < 4; ++j) {
    const int i = tid + j * 256;
    comb[b * HH + i] = attn[b * HH + i] + (hv[j] - mean) * rstd * gamma[i] + beta[i];
  }
}

// ---------------------------------------------------------------------------
extern "C" void kernel_launch(void* const* d_in, const int* in_sizes, int n_in,
                              void* d_out, int out_size, void* d_ws, size_t ws_size,
                              hipStream_t stream) {
  const float* x       = (const float*)d_in[0];
  const float* Wq_proj = (const float*)d_in[1];
  const float* bq      = (const float*)d_in[2];
  const float* Wq      = (const float*)d_in[3];
  const float* Wk      = (const float*)d_in[4];
  const float* Wv      = (const float*)d_in[5];
  const float* W1      = (const float*)d_in[6];
  const float* b1      = (const float*)d_in[7];
  const float* W2      = (const float*)d_in[8];
  const float* b2      = (const float*)d_in[9];
  const float* gamma   = (const float*)d_in[10];
  const float* beta    = (const float*)d_in[11];
  float* out = (float*)d_out;

  float* ws     = (float*)d_ws;
  float* qproj  = ws;             // [4][H]
  float* qlast  = ws + 4096;      // [4][H]
  float* qtil   = ws + 8192;      // [4][D]
  float* scores = ws + 12288;     // [4][S] (becomes weights in place)
  float* cvec   = ws + 20480;     // [4][D]
  float* attn   = ws + 24576;     // [4][H]
  float* hin    = ws + 28672;     // [4][2H]
  float* comb   = ws + 36864;     // [4][H]

  // 1) qproj = x[:, S-1, :] @ Wq_proj + bq   (A rows strided by S*D)
  wmma_gemm16<false><<<dim3(HH / 16), 128, 0, stream>>>(
      x + (size_t)(SS - 1) * DD, (long)SS * DD, BN, Wq_proj, HH, DD, bq, qproj, HH);
  // 2) qlast = qproj @ Wq
  wmma_gemm16<false><<<dim3(HH / 16), 128, 0, stream>>>(
      qproj, HH, BN, Wq, HH, HH, nullptr, qlast, HH);
  // 3) qtil = qlast @ Wk^T   (scores_s = qtil . x_s; K never materialized)
  wmma_gemm16<true><<<dim3(DD / 16), 128, 0, stream>>>(
      qlast, HH, BN, Wk, HH, HH, nullptr, qtil, DD);
  // 4) scores over all s (stream x from HBM once)
  scores_k<<<dim3(SS / 8, BN), 256, 0, stream>>>(x, qtil, scores);
  // 5) softmax per batch (+ zero cvec)
  softmax_zero<<<dim3(BN), 256, 0, stream>>>(scores, cvec);
  // 6) cvec = weights @ x  (x re-read from L2)
  wsum_k<<<dim3(16, BN), 256, 0, stream>>>(x, scores, cvec);
  // 7) attn = cvec @ Wv    (V never materialized)
  wmma_gemm16<false><<<dim3(HH / 16), 128, 0, stream>>>(
      cvec, DD, BN, Wv, HH, DD, nullptr, attn, HH);
  // 8) hin = attn @ W1 + b1
  wmma_gemm16<false><<<dim3(FF / 16), 128, 0, stream>>>(
      attn, HH, BN, W1, FF, HH, b1, hin, FF);
  // 9) GLU + LayerNorm + residual
  glu_ln<<<dim3(BN), 256, 0, stream>>>(hin, attn, gamma, beta, comb);
  // 10) out = comb @ W2 + b2  -> [4, 1024]
  wmma_gemm16<false><<<dim3(OO / 16), 128, 0, stream>>>(
      comb, HH, BN, W2, OO, HH, b2, out, OO);
}